// HungarianMatcherDynamicK_12249246728996
// MI455X (gfx1250) — compile-verified
//
#include <hip/hip_runtime.h>
#include <math.h>

#define BB 16
#define QQ 4096
#define GG 128
#define CC 80

typedef __attribute__((ext_vector_type(2))) float v2f;
typedef __attribute__((ext_vector_type(8))) float v8f;

// ---------------------------------------------------------------------------
// Stage 0a: per-GT geometry precompute
// gdata layout (12 floats per gt): [ax0, ay0, ax1, ay1, cx, cy, area, thr_box, radius, 0,0,0]
// ---------------------------------------------------------------------------
__global__ void k_gt_pre(const float* __restrict__ gtb, float* __restrict__ gdata) {
    int idx = blockIdx.x * blockDim.x + threadIdx.x;
    if (idx >= BB * GG) return;
    const float* c = gtb + (size_t)idx * 8;
    float x0 = c[0], y0 = c[1], x1 = c[2], y1 = c[3];
    float x2 = c[4], y2 = c[5], x3 = c[6], y3 = c[7];
    float cx = (x0 + x1 + x2 + x3) * 0.25f;
    float cy = (y0 + y1 + y2 + y3) * 0.25f;
    float w = sqrtf((x0 - x1) * (x0 - x1) + (y0 - y1) * (y0 - y1));
    float h = sqrtf((x1 - x2) * (x1 - x2) + (y1 - y2) * (y1 - y2));
    float v1x = x1 - x0, v1y = y1 - y0, v2x = x2 - x0, v2y = y2 - y0;
    float garea = fabsf(v1x * v2y - v1y * v2x) * 0.5f;
    float mx = fmaxf(fmaxf(x0, x1), fmaxf(x2, x3));
    float mnx = fminf(fminf(x0, x1), fminf(x2, x3));
    float my = fmaxf(fmaxf(y0, y1), fmaxf(y2, y3));
    float mny = fminf(fminf(y0, y1), fminf(y2, y3));
    float diag = sqrtf((mx - mnx) * (mx - mnx) + (my - mny) * (my - mny));
    float radius = diag / 32.0f;
    float thr = garea / radius;
    float* o = gdata + (size_t)idx * 12;
    o[0] = cx - 0.5f * w;
    o[1] = cy - 0.5f * h;
    o[2] = cx + 0.5f * w;
    o[3] = cy + 0.5f * h;
    o[4] = cx;
    o[5] = cy;
    o[6] = w * h;
    o[7] = thr;
    o[8] = radius;
    o[9] = 0.f; o[10] = 0.f; o[11] = 0.f;
}

// ---------------------------------------------------------------------------
// Stage 0b: per-query foreground mask via WMMA pairwise distances.
// One wave per 16-query band; 8 unrolled V_WMMA_F32_16X16X4_F32 cover all 128
// gts; per-row "any" reduced across the 16 column lanes with shfl_xor
// butterflies (stay within each 16-lane half on wave32).
// ---------------------------------------------------------------------------
__global__ void k_q_fg(const float* __restrict__ pbx, const float* __restrict__ gdata,
                       float* __restrict__ qfg) {
    __shared__ float sqcx[128], sqcy[128];
    __shared__ float sgcx[GG], sgcy[GG], sgthr[GG];
    int t = threadIdx.x;
    int b = blockIdx.z;
    int q0 = blockIdx.x * 128;

    if (t < 128) {
        const float* c = pbx + ((size_t)b * QQ + q0 + t) * 8;
        sqcx[t] = (c[0] + c[2] + c[4] + c[6]) * 0.25f;
        sqcy[t] = (c[1] + c[3] + c[5] + c[7]) * 0.25f;
    } else {
        int g = t - 128;
        const float* gd = gdata + ((size_t)b * GG + g) * 12;
        sgcx[g] = gd[4];
        sgcy[g] = gd[5];
        sgthr[g] = gd[7];
    }
    __syncthreads();

    int lane = t & 31;
    int wv = t >> 5;       // 16-query band
    int half = lane >> 4;  // 0/1
    int lm = lane & 15;

    float qcx = sqcx[wv * 16 + lm];
    float qcy = sqcy[wv * 16 + lm];
    v2f A;
    A.x = half ? (qcx * qcx + qcy * qcy) : qcx;
    A.y = half ? 1.0f : qcy;

    unsigned rowmask = 0u;
#pragma unroll
    for (int gt = 0; gt < 8; ++gt) {
        int g = gt * 16 + lm;
        float gx = sgcx[g], gy = sgcy[g];
        v2f Bv;
        Bv.x = half ? 1.0f : (-2.0f * gx);
        Bv.y = half ? (gx * gx + gy * gy) : (-2.0f * gy);
        v8f d2 = {};
        d2 = __builtin_amdgcn_wmma_f32_16x16x4_f32(false, A, false, Bv,
                                                   (short)0, d2, false, false);
        float thr = sgthr[g];
#pragma unroll
        for (int r = 0; r < 8; ++r) {
            if (sqrtf(fmaxf(d2[r], 0.0f)) <= thr) rowmask |= (1u << r);
        }
    }
    // OR across the 16 lanes of each half (xor masks 1,2,4,8 stay in-half)
    rowmask |= (unsigned)__shfl_xor((int)rowmask, 1);
    rowmask |= (unsigned)__shfl_xor((int)rowmask, 2);
    rowmask |= (unsigned)__shfl_xor((int)rowmask, 4);
    rowmask |= (unsigned)__shfl_xor((int)rowmask, 8);
    if (lm == 0) {
#pragma unroll
        for (int r = 0; r < 8; ++r) {
            int q = q0 + wv * 16 + r + 8 * half;
            qfg[(size_t)b * QQ + q] = ((rowmask >> r) & 1u) ? 1.0f : 0.0f;
        }
    }
}

// ---------------------------------------------------------------------------
// Stage 1: pairwise cost / iou.  One wave per 16x16 (q,g) tile.
// Query corners staged to LDS with global_load_async_to_lds_b128 (ASYNCcnt),
// pairwise center distance d2 via V_WMMA_F32_16X16X4_F32:
//   A row (q):  [qx, qy, qx^2+qy^2, 1]         (16x4)
//   B col (g):  [-2gx, -2gy, 1, gx^2+gy^2]     (4x16)
//   D = A*B = (qx-gx)^2 + (qy-gy)^2            (16x16)
// ---------------------------------------------------------------------------
__global__ void k_pair(const float* __restrict__ logits, const float* __restrict__ pbx,
                       const float* __restrict__ gtb, const int* __restrict__ glab,
                       const float* __restrict__ gdata, const float* __restrict__ qfg,
                       float* __restrict__ out_match, float* __restrict__ out_cost,
                       float* __restrict__ out_iou) {
    __shared__ __attribute__((aligned(16))) float sqc[128][8];  // query corners (async-staged)
    __shared__ float sqa[128][4];    // query pseudo-aabb
    __shared__ float sqcen[128][2];  // query center
    __shared__ float sqarea[128];
    __shared__ float sqfgm[128];
    __shared__ float sgc[16][8];     // gt corners
    __shared__ float sga[16][4];
    __shared__ float sgcen[16][2];
    __shared__ float sgarea[16];
    __shared__ int   sglb[16];

    int t = threadIdx.x;
    int b = blockIdx.z;
    int q0 = blockIdx.x * 128;
    int g0 = blockIdx.y * 16;

    // Prefetch the logits rows this block will gather from.
    if (t < 128) {
        const float* lrow = logits + ((size_t)b * QQ + q0 + t) * CC;
        __builtin_prefetch(lrow, 0, 0);
        __builtin_prefetch(lrow + 64, 0, 0);
    }

    // Async-stage 128 query corner rows (4KB) into LDS: 16B per thread.
    {
        unsigned ldsoff = (unsigned)(uintptr_t)(&sqc[0][0]) + (unsigned)(t * 16);
        unsigned long long ga =
            (unsigned long long)(uintptr_t)(pbx + ((size_t)b * QQ + q0) * 8) +
            (unsigned long long)(t * 16);
        asm volatile("global_load_async_to_lds_b128 %0, %1, off"
                     :: "v"(ldsoff), "v"(ga) : "memory");
    }

    if (t >= 128 && t < 144) {
        int g = t - 128;
        const float* gd = gdata + ((size_t)b * GG + g0 + g) * 12;
        sga[g][0] = gd[0]; sga[g][1] = gd[1]; sga[g][2] = gd[2]; sga[g][3] = gd[3];
        sgcen[g][0] = gd[4]; sgcen[g][1] = gd[5];
        sgarea[g] = gd[6];
        sglb[g] = glab[b * GG + g0 + g];
        const float* gc = gtb + ((size_t)b * GG + g0 + g) * 8;
        for (int k = 0; k < 8; ++k) sgc[g][k] = gc[k];
    }
    asm volatile("s_wait_asynccnt 0x0" ::: "memory");
    __syncthreads();

    // Derived per-query values from the async-staged corners.
    if (t < 128) {
        float x0 = sqc[t][0], y0 = sqc[t][1], x1 = sqc[t][2], y1 = sqc[t][3];
        float x2 = sqc[t][4], y2 = sqc[t][5], x3 = sqc[t][6], y3 = sqc[t][7];
        float cx = (x0 + x1 + x2 + x3) * 0.25f;
        float cy = (y0 + y1 + y2 + y3) * 0.25f;
        float w = sqrtf((x0 - x1) * (x0 - x1) + (y0 - y1) * (y0 - y1));
        float h = sqrtf((x1 - x2) * (x1 - x2) + (y1 - y2) * (y1 - y2));
        sqa[t][0] = cx - 0.5f * w; sqa[t][1] = cy - 0.5f * h;
        sqa[t][2] = cx + 0.5f * w; sqa[t][3] = cy + 0.5f * h;
        sqcen[t][0] = cx; sqcen[t][1] = cy;
        sqarea[t] = w * h;
        sqfgm[t] = qfg[(size_t)b * QQ + q0 + t];
    }
    __syncthreads();

    int lane = t & 31;
    int wv = t >> 5;       // 0..7 -> which 16-row band of the 128 queries
    int half = lane >> 4;  // 0/1
    int lm = lane & 15;

    // Build A operand: row m = lm; lanes<16 hold (K0,K1), lanes>=16 hold (K2,K3)
    float qcx = sqcen[wv * 16 + lm][0];
    float qcy = sqcen[wv * 16 + lm][1];
    v2f A;
    A.x = half ? (qcx * qcx + qcy * qcy) : qcx;
    A.y = half ? 1.0f : qcy;
    // Build B operand: col n = lm
    float gcx = sgcen[lm][0];
    float gcy = sgcen[lm][1];
    v2f Bv;
    Bv.x = half ? 1.0f : (-2.0f * gcx);
    Bv.y = half ? (gcx * gcx + gcy * gcy) : (-2.0f * gcy);

    v8f d2t = {};
    d2t = __builtin_amdgcn_wmma_f32_16x16x4_f32(
        /*neg_a=*/false, A, /*neg_b=*/false, Bv,
        /*c_mod=*/(short)0, d2t, /*reuse_a=*/false, /*reuse_b=*/false);

    // Epilogue: each lane owns 8 rows (M = r + 8*half) of column N = lm.
#pragma unroll
    for (int r = 0; r < 8; ++r) {
        int qi = wv * 16 + r + 8 * half;
        int gi = lm;
        int q = q0 + qi;
        int g = g0 + gi;
        // AABB IoU
        float ltx = fmaxf(sqa[qi][0], sga[gi][0]);
        float lty = fmaxf(sqa[qi][1], sga[gi][1]);
        float rbx = fminf(sqa[qi][2], sga[gi][2]);
        float rby = fminf(sqa[qi][3], sga[gi][3]);
        float iw = fmaxf(rbx - ltx, 0.0f);
        float ih = fmaxf(rby - lty, 0.0f);
        float inter = iw * ih;
        float iou = inter / (sqarea[qi] + sgarea[gi] - inter + 1e-8f);
        // DIoU with WMMA-computed d2
        float ex = fmaxf(sqa[qi][2], sga[gi][2]) - fminf(sqa[qi][0], sga[gi][0]);
        float ey = fmaxf(sqa[qi][3], sga[gi][3]) - fminf(sqa[qi][1], sga[gi][1]);
        float c2 = ex * ex + ey * ey + 1e-8f;
        float diou = iou - d2t[r] / c2;
        // L1 corner cost
        float cb = 0.0f;
#pragma unroll
        for (int k = 0; k < 8; ++k) cb += fabsf(sqc[qi][k] - sgc[gi][k]);
        // focal class cost (gathered at gt label)
        int lab = sglb[gi];
        float lg = logits[((size_t)b * QQ + q) * CC + lab];
        float p = 1.0f / (1.0f + expf(-lg));
        float pos = 0.25f * (1.0f - p) * (1.0f - p) * (-logf(p + 1e-8f));
        float neg = 0.75f * p * p * (-logf(1.0f - p + 1e-8f));
        // cost = 5*bbox + 2*class + 2*diou (ref: COST_GIOU * -cost_giou = 2*diou)
        float cost = 5.0f * cb + 2.0f * (pos - neg) + 2.0f * diou;
        if (sqfgm[qi] < 0.5f) cost += 10000.0f;
        size_t o = ((size_t)(b * QQ + q)) * GG + g;
        out_match[o] = 0.0f;
        out_cost[o] = cost;
        out_iou[o] = iou;
    }
}

// ---------------------------------------------------------------------------
// Stage 2: per (b,g) column: dyn_k = clip(int(sum top5 iou),1,Q)  (<=5),
// then set matching=1 for the dyn_k smallest (cost, q) lexicographic pairs
// (== rank-from-stable-double-argsort < dyn_k).
// ---------------------------------------------------------------------------
__device__ inline void ins_desc5(float* a, float v) {
    if (v <= a[4]) return;
    a[4] = v;
    for (int i = 4; i > 0 && a[i] > a[i - 1]; --i) {
        float tmp = a[i]; a[i] = a[i - 1]; a[i - 1] = tmp;
    }
}
__device__ inline void ins_lex5(float* c, int* q, float v, int qi) {
    if (v > c[4] || (v == c[4] && qi >= q[4])) return;
    c[4] = v; q[4] = qi;
    for (int i = 4; i > 0; --i) {
        if (c[i] < c[i - 1] || (c[i] == c[i - 1] && q[i] < q[i - 1])) {
            float tc = c[i]; c[i] = c[i - 1]; c[i - 1] = tc;
            int tq = q[i]; q[i] = q[i - 1]; q[i - 1] = tq;
        } else break;
    }
}

__global__ void k_col(const float* __restrict__ cost, const float* __restrict__ iou,
                      float* __restrict__ match) {
    int bg = blockIdx.x;
    int b = bg / GG, g = bg % GG;
    const float* colI = iou + ((size_t)b * QQ) * GG + g;
    const float* colC = cost + ((size_t)b * QQ) * GG + g;
    float* colM = match + ((size_t)b * QQ) * GG + g;
    int t = threadIdx.x;

    __shared__ float sI[256 * 5];
    __shared__ float sC[256 * 5];
    __shared__ int sQ[256 * 5];
    __shared__ int resQ[5];
    __shared__ int dynk;

    float ti[5] = {-1e30f, -1e30f, -1e30f, -1e30f, -1e30f};
    float lc[5] = {1e38f, 1e38f, 1e38f, 1e38f, 1e38f};
    int li[5] = {QQ, QQ, QQ, QQ, QQ};
    for (int q = t; q < QQ; q += 256) {
        ins_desc5(ti, colI[(size_t)q * GG]);
        ins_lex5(lc, li, colC[(size_t)q * GG], q);
    }
    for (int j = 0; j < 5; ++j) {
        sI[t * 5 + j] = ti[j];
        sC[t * 5 + j] = lc[j];
        sQ[t * 5 + j] = li[j];
    }
    __syncthreads();
    if (t == 0) {
        float g5[5] = {-1e30f, -1e30f, -1e30f, -1e30f, -1e30f};
        for (int i = 0; i < 256 * 5; ++i) ins_desc5(g5, sI[i]);
        float s = g5[0] + g5[1] + g5[2] + g5[3] + g5[4];
        int k = (int)s;
        if (k < 1) k = 1;
        if (k > 5) k = 5;
        dynk = k;
        float bc5[5] = {1e38f, 1e38f, 1e38f, 1e38f, 1e38f};
        int bq5[5] = {QQ, QQ, QQ, QQ, QQ};
        for (int i = 0; i < 256 * 5; ++i) ins_lex5(bc5, bq5, sC[i], sQ[i]);
        for (int j = 0; j < 5; ++j) resQ[j] = bq5[j];
    }
    __syncthreads();
    if (t < dynk && resQ[t] < QQ) colM[(size_t)resQ[t] * GG] = 1.0f;
}

// ---------------------------------------------------------------------------
// Stage 3: initial dedup — rows matched to >1 gts become one-hot at row argmin.
// ---------------------------------------------------------------------------
__global__ void k_dedup(const float* __restrict__ cost, float* __restrict__ match) {
    int idx = blockIdx.x * blockDim.x + threadIdx.x;
    if (idx >= BB * QQ) return;
    size_t base = (size_t)idx * GG;
    int cnt = 0;
    for (int g = 0; g < GG; ++g) cnt += (match[base + g] > 0.5f) ? 1 : 0;
    if (cnt <= 1) return;
    int amin = 0;
    float mv = cost[base];
    for (int g = 1; g < GG; ++g) {
        float v = cost[base + g];
        if (v < mv) { mv = v; amin = g; }
    }
    for (int g = 0; g < GG; ++g) match[base + g] = (g == amin) ? 1.0f : 0.0f;
}

// ---------------------------------------------------------------------------
// Stage 4: assignment while-loop.  One block per batch.
// adjusted c[q,g] = cost[q,g] + 1e5 * acc[q]; acc[q]++ each iter row q matched.
// Row-wise dedup argmin is invariant to the row-constant penalty.
// ---------------------------------------------------------------------------
__global__ void k_iter(const float* __restrict__ cost, float* __restrict__ match) {
    int b = blockIdx.x;
    int t = threadIdx.x;
    const float* C = cost + (size_t)b * QQ * GG;
    float* M = match + (size_t)b * QQ * GG;

    __shared__ int rowcnt[QQ];
    __shared__ int accp[QQ];
    __shared__ int colcnt[GG];
    __shared__ int anyUnm;

    for (int g = t; g < GG; g += 256) colcnt[g] = 0;
    __syncthreads();
    for (int q = t; q < QQ; q += 256) {
        accp[q] = 0;
        int c = 0;
        size_t rb = (size_t)q * GG;
        for (int g = 0; g < GG; ++g)
            if (M[rb + g] > 0.5f) { c++; atomicAdd(&colcnt[g], 1); }
        rowcnt[q] = c;
    }
    __syncthreads();

    for (int it = 0; it < GG; ++it) {
        if (t == 0) anyUnm = 0;
        __syncthreads();
        for (int g = t; g < GG; g += 256)
            if (colcnt[g] == 0) anyUnm = 1;
        __syncthreads();
        if (!anyUnm) break;
        // penalty for currently matched rows
        for (int q = t; q < QQ; q += 256)
            if (rowcnt[q] > 0) accp[q]++;
        __syncthreads();
        // assign each unmatched column to its argmin row
        if (t < GG && colcnt[t] == 0) {
            int g = t;
            float mv = 1e38f;
            int amin = 0;
            for (int q = 0; q < QQ; ++q) {
                float v = C[(size_t)q * GG + g] + 100000.0f * (float)accp[q];
                if (v < mv) { mv = v; amin = q; }
            }
            M[(size_t)amin * GG + g] = 1.0f;
            atomicAdd(&rowcnt[amin], 1);
            colcnt[g] = 1;
        }
        __syncthreads();
        // dedup multi-matched rows to one-hot at row argmin
        for (int q = t; q < QQ; q += 256) {
            if (rowcnt[q] > 1) {
                size_t rb = (size_t)q * GG;
                float mv = C[rb];
                int amin = 0;
                for (int g = 1; g < GG; ++g) {
                    float v = C[rb + g];
                    if (v < mv) { mv = v; amin = g; }
                }
                for (int g = 0; g < GG; ++g) {
                    bool on = M[rb + g] > 0.5f;
                    if (g == amin) {
                        if (!on) { M[rb + g] = 1.0f; atomicAdd(&colcnt[g], 1); }
                    } else if (on) {
                        M[rb + g] = 0.0f;
                        atomicSub(&colcnt[g], 1);
                    }
                }
                rowcnt[q] = 1;
            }
        }
        __syncthreads();
    }
}

// ---------------------------------------------------------------------------
extern "C" void kernel_launch(void* const* d_in, const int* in_sizes, int n_in,
                              void* d_out, int out_size, void* d_ws, size_t ws_size,
                              hipStream_t stream) {
    (void)in_sizes; (void)n_in; (void)out_size; (void)ws_size;
    const float* logits = (const float*)d_in[0];  // (B,Q,C)
    const float* pbx = (const float*)d_in[1];     // (B,Q,8)
    const float* gtb = (const float*)d_in[2];     // (B,G,8)
    const int* glab = (const int*)d_in[3];        // (B,G)

    float* out = (float*)d_out;
    size_t N = (size_t)BB * QQ * GG;
    float* out_match = out;
    float* out_cost = out + N;
    float* out_iou = out + 2 * N;

    float* gdata = (float*)d_ws;                      // BB*GG*12 floats
    float* qfg = gdata + (size_t)BB * GG * 12;        // BB*QQ floats

    k_gt_pre<<<(BB * GG + 255) / 256, 256, 0, stream>>>(gtb, gdata);
    dim3 fgrid(QQ / 128, 1, BB);
    k_q_fg<<<fgrid, 256, 0, stream>>>(pbx, gdata, qfg);
    dim3 grid(QQ / 128, GG / 16, BB);
    k_pair<<<grid, 256, 0, stream>>>(logits, pbx, gtb, glab, gdata, qfg,
                                     out_match, out_cost, out_iou);
    k_col<<<BB * GG, 256, 0, stream>>>(out_cost, out_iou, out_match);
    k_dedup<<<(BB * QQ + 255) / 256, 256, 0, stream>>>(out_cost, out_match);
    k_iter<<<BB, 256, 0, stream>>>(out_cost, out_match);
}